// RiemannianBatchNorm_57853209477639
// MI455X (gfx1250) — compile-verified
//
#include <hip/hip_runtime.h>
#include <math.h>

typedef __attribute__((ext_vector_type(4))) float v4f;
typedef __attribute__((ext_vector_type(2))) float v2f;
typedef __attribute__((ext_vector_type(8))) float v8f;

#define DDIM 128
#define WPB 8            // waves per block (256 threads, wave32)
#define GRID_BLKS 1024

// ws layout (floats):
// [0..127]   acc vector accumulator
// [128..255] mu (current Frechet mean)
// [256..383] t  = on_manifold = exp0(mean_param)
// [384..511] r  = -mobius_add(t, -mu)
// [512] scalar accumulator (variance)
// [513] mu2   [514] t2   [515] r2   [516] var   [517] sfac

__device__ __forceinline__ float wsum(float v) {
    v += __shfl_xor(v, 16);
    v += __shfl_xor(v, 8);
    v += __shfl_xor(v, 4);
    v += __shfl_xor(v, 2);
    v += __shfl_xor(v, 1);
    return v;
}

__device__ __forceinline__ float dot4(v4f a, v4f b) {
    return wsum(a.x * b.x + a.y * b.y + a.z * b.z + a.w * b.w);
}

__device__ __forceinline__ float artanh_(float z) {
    // z in [0, 1-1e-6]
    return 0.5f * logf((1.f + z) / (1.f - z));
}

// ---------------- tiny single-wave kernels ----------------

__global__ void k_init(const float* __restrict__ mp, float* __restrict__ ws) {
    int lane = threadIdx.x;                 // 32 threads
    v4f v = ((const v4f*)mp)[lane];
    float n2 = dot4(v, v);
    float n = sqrtf(fmaxf(n2, 1e-12f));
    float c = tanhf(n) / n;
    v4f t = v * c;                          // t = exp0(mean_param)
    ((v4f*)(ws + 256))[lane] = t;
    float t2 = dot4(t, t);
    if (lane == 0) { ws[514] = t2; ws[512] = 0.f; ws[513] = 0.f; }
    v4f z = {0.f, 0.f, 0.f, 0.f};
    ((v4f*)ws)[lane] = z;                   // zero vector accumulator
}

__global__ void k_update(float* __restrict__ ws, int N, int mode) {
    int lane = threadIdx.x;                 // 32 threads
    v4f acc = ((v4f*)ws)[lane];
    v4f w = acc * (1.f / (float)N);         // mean tangent
    v4f mu_new;
    if (mode == 0) {                        // mu = exp0(w)
        float n2 = dot4(w, w);
        float n = sqrtf(fmaxf(n2, 1e-12f));
        float c = tanhf(n) / n;
        mu_new = w * c;
    } else {                                // mu = expmap(mu, w)
        v4f mu = ((v4f*)(ws + 128))[lane];
        float mu2 = ws[513];
        float n2 = dot4(w, w);
        float n = sqrtf(fmaxf(n2, 1e-12f));
        float arg = fminf(n / fmaxf(1.f - mu2, 1e-12f), 15.f);
        float c = tanhf(arg) / n;
        v4f sec = w * c;
        float y2 = dot4(sec, sec);
        float xy = dot4(mu, sec);
        float cA = 1.f + 2.f * xy + y2;
        float cB = 1.f - mu2;
        float den = fmaxf(1.f + 2.f * xy + mu2 * y2, 1e-12f);
        mu_new = (mu * cA + sec * cB) / den;
    }
    ((v4f*)(ws + 128))[lane] = mu_new;
    float mu2n = dot4(mu_new, mu_new);
    if (lane == 0) ws[513] = mu2n;
    v4f z = {0.f, 0.f, 0.f, 0.f};
    ((v4f*)ws)[lane] = z;                   // re-zero accumulator for next pass
}

__global__ void k_prep(float* __restrict__ ws, const float* __restrict__ var_param, int N) {
    int lane = threadIdx.x;                 // 32 threads
    v4f t  = ((v4f*)(ws + 256))[lane];
    v4f mu = ((v4f*)(ws + 128))[lane];
    float t2 = ws[514];
    float mu2 = ws[513];
    float xy = -dot4(t, mu);                // <t, -mu>
    float cA = 1.f + 2.f * xy + mu2;
    float cB = 1.f - t2;
    float den = fmaxf(1.f + 2.f * xy + t2 * mu2, 1e-12f);
    v4f e = (t * cA + (-mu) * cB) / den;    // mobius_add(t, -mu)
    v4f r = -e;
    ((v4f*)(ws + 384))[lane] = r;
    float r2 = dot4(r, r);
    float var = ws[512] / (float)N;
    float sfac = (fmaxf(1.f - t2, 1e-12f) / fmaxf(1.f - mu2, 1e-12f))
               * sqrtf(var_param[0] / (var + 1e-6f));
    if (lane == 0) { ws[515] = r2; ws[516] = var; ws[517] = sfac; }
}

// ---------------- streaming reduction pass ----------------
// mode 0: acc += log0(x)        mode 1: acc += logmap(mu, x)
// mode 2: acc_scalar += dist(x, mu)^2

__global__ __launch_bounds__(256) void k_reduce(const float* __restrict__ x,
                                                float* __restrict__ ws,
                                                int N, int mode) {
    __shared__ float lds[WPB * DDIM];
    const int lane = threadIdx.x & 31;
    const int wave = threadIdx.x >> 5;
    const int gwave = blockIdx.x * WPB + wave;
    const int nwaves = gridDim.x * WPB;

    const float mu2 = ws[513];
    const float invlam = fmaxf(1.f - mu2, 1e-12f);   // 2/lambda_mu
    v4f mu = {0.f, 0.f, 0.f, 0.f};
    if (mode != 0) mu = ((const v4f*)(ws + 128))[lane];
    v4f nmu = -mu;

    v4f accv = {0.f, 0.f, 0.f, 0.f};
    float accs = 0.f;

    for (int row = gwave; row < N; row += nwaves) {
        const v4f* rp = (const v4f*)(x + (size_t)row * DDIM);
        __builtin_prefetch(x + (size_t)(row + nwaves) * DDIM + 4 * lane, 0, 0);
        v4f xv = rp[lane];
        if (mode == 0) {
            float x2 = dot4(xv, xv);
            float n = sqrtf(fmaxf(x2, 1e-12f));
            float c = artanh_(fminf(n, 1.f - 1e-6f)) / n;
            accv += xv * c;
        } else {
            float x2 = dot4(xv, xv);
            float mux = dot4(mu, xv);
            float ab = -mux;
            float cA = 1.f + 2.f * ab + x2;
            float cB = 1.f - mu2;
            float den = fmaxf(1.f + 2.f * ab + mu2 * x2, 1e-12f);
            v4f m = (nmu * cA + xv * cB) / den;      // mobius_add(-mu, x)
            float m2 = dot4(m, m);
            float n = sqrtf(fmaxf(m2, 1e-12f));
            float nc = fminf(n, 1.f - 1e-6f);
            if (mode == 1) {
                float c = invlam * artanh_(nc) / n;
                accv += m * c;
            } else {
                float dd = 2.f * artanh_(nc);        // same on all lanes
                accs += dd * dd;
            }
        }
    }

    if (mode == 2) {
        if (lane == 0) lds[wave] = accs;             // accs is wave-uniform
        __syncthreads();
        if (threadIdx.x == 0) {
            float s = 0.f;
            for (int w = 0; w < WPB; ++w) s += lds[w];
            atomicAdd(ws + 512, s);
        }
    } else {
        ((v4f*)(lds + wave * DDIM))[lane] = accv;    // M[8][128] wave partials
        __syncthreads();
        if (wave == 0) {
            // Column-sum M[8][128] with fp32 WMMA: D = ones(16x4) * B(4x16) + C,
            // chained twice to cover 8 rows. Column = lane%16 for B and D.
            const int cin = lane & 15;
            const int rsel = lane >> 4;              // 0 or 1
            v2f ones; ones.x = 1.f; ones.y = 1.f;
            for (int g = 0; g < 8; ++g) {
                int c = g * 16 + cin;
                v2f b0, b1;
                b0.x = lds[(rsel + 0) * DDIM + c];
                b0.y = lds[(rsel + 2) * DDIM + c];
                b1.x = lds[(rsel + 4) * DDIM + c];
                b1.y = lds[(rsel + 6) * DDIM + c];
                v8f cc = {};
                cc = __builtin_amdgcn_wmma_f32_16x16x4_f32(
                        false, ones, false, b0, (short)0, cc, false, false);
                cc = __builtin_amdgcn_wmma_f32_16x16x4_f32(
                        false, ones, false, b1, (short)0, cc, false, false);
                if (lane < 16) atomicAdd(ws + c, cc[0]);
            }
        }
    }
}

// ---------------- final elementwise pass ----------------

__global__ __launch_bounds__(256) void k_final(const float* __restrict__ x,
                                               const float* __restrict__ ws,
                                               float* __restrict__ out, int N) {
    const int lane = threadIdx.x & 31;
    const int wave = threadIdx.x >> 5;
    const int gwave = blockIdx.x * WPB + wave;
    const int nwaves = gridDim.x * WPB;

    v4f mu = ((const v4f*)(ws + 128))[lane];
    v4f t  = ((const v4f*)(ws + 256))[lane];
    v4f r  = ((const v4f*)(ws + 384))[lane];
    const float mu2 = ws[513], t2 = ws[514], r2 = ws[515], sfac = ws[517];
    const float invlam_mu = fmaxf(1.f - mu2, 1e-12f);
    const float invlam_t  = fmaxf(1.f - t2, 1e-12f);
    v4f nmu = -mu;

    for (int row = gwave; row < N; row += nwaves) {
        __builtin_prefetch(x + (size_t)(row + nwaves) * DDIM + 4 * lane, 0, 0);
        v4f xv = ((const v4f*)(x + (size_t)row * DDIM))[lane];
        // w = logmap(mu, x)
        float x2 = dot4(xv, xv);
        float mux = dot4(mu, xv);
        float ab = -mux;
        float cA = 1.f + 2.f * ab + x2;
        float cB = 1.f - mu2;
        float den = fmaxf(1.f + 2.f * ab + mu2 * x2, 1e-12f);
        v4f m = (nmu * cA + xv * cB) / den;
        float m2 = dot4(m, m);
        float n = sqrtf(fmaxf(m2, 1e-12f));
        float c0 = invlam_mu * artanh_(fminf(n, 1.f - 1e-6f)) / n;
        v4f w = m * c0;
        // q = mobius_add(-mu, w)
        float w2 = dot4(w, w);
        float muw = dot4(mu, w);
        ab = -muw;
        cA = 1.f + 2.f * ab + w2; cB = 1.f - mu2;
        den = fmaxf(1.f + 2.f * ab + mu2 * w2, 1e-12f);
        v4f q = (nmu * cA + w * cB) / den;
        // p = mobius_add(t, q)
        float q2 = dot4(q, q);
        float tq = dot4(t, q);
        cA = 1.f + 2.f * tq + q2; cB = 1.f - t2;
        den = fmaxf(1.f + 2.f * tq + t2 * q2, 1e-12f);
        v4f p = (t * cA + q * cB) / den;
        // g = mobius_add(r, p)  -> gyr[t,-mu]w
        float p2 = dot4(p, p);
        float rp = dot4(r, p);
        cA = 1.f + 2.f * rp + p2; cB = 1.f - r2;
        den = fmaxf(1.f + 2.f * rp + r2 * p2, 1e-12f);
        v4f g = (r * cA + p * cB) / den;
        // scale (lam ratio * sqrt(var_param/(var+eps))) then expmap(t, .)
        v4f s = g * sfac;
        float s2 = dot4(s, s);
        float sn = sqrtf(fmaxf(s2, 1e-12f));
        float arg = fminf(sn / invlam_t, 15.f);
        float cs = tanhf(arg) / sn;
        v4f sec = s * cs;
        float sec2 = dot4(sec, sec);
        float tsec = dot4(t, sec);
        cA = 1.f + 2.f * tsec + sec2; cB = 1.f - t2;
        den = fmaxf(1.f + 2.f * tsec + t2 * sec2, 1e-12f);
        v4f o = (t * cA + sec * cB) / den;
        __builtin_nontemporal_store(o, (v4f*)(out + (size_t)row * DDIM) + lane);
    }
}

extern "C" void kernel_launch(void* const* d_in, const int* in_sizes, int n_in,
                              void* d_out, int out_size, void* d_ws, size_t ws_size,
                              hipStream_t stream) {
    const float* x = (const float*)d_in[0];
    const float* mean_param = (const float*)d_in[1];
    const float* var_param = (const float*)d_in[2];
    float* out = (float*)d_out;
    float* ws = (float*)d_ws;
    const int N = in_sizes[0] / DDIM;

    k_init<<<1, 32, 0, stream>>>(mean_param, ws);
    k_reduce<<<GRID_BLKS, 256, 0, stream>>>(x, ws, N, 0);
    k_update<<<1, 32, 0, stream>>>(ws, N, 0);
    for (int it = 0; it < 15; ++it) {
        k_reduce<<<GRID_BLKS, 256, 0, stream>>>(x, ws, N, 1);
        k_update<<<1, 32, 0, stream>>>(ws, N, 1);
    }
    k_reduce<<<GRID_BLKS, 256, 0, stream>>>(x, ws, N, 2);   // variance
    k_prep<<<1, 32, 0, stream>>>(ws, var_param, N);
    k_final<<<GRID_BLKS, 256, 0, stream>>>(x, ws, out, N);
}